// BS_4355096838362
// MI455X (gfx1250) — compile-verified
//
#include <hip/hip_runtime.h>
#include <stdint.h>

// ---------------------------------------------------------------------------
// B-spline basis (DF=16, degree 3) on MI455X.
// Memory-bound: 256MB out + 16MB in @ 23.3 TB/s  => ~11.7us floor.
// Compute reduced to ~25 VALU ops/t via local de Boor with constexpr
// per-span reciprocal-denominator tables (matches reference's +1e-15 eps).
// Output staged in LDS (conflict-free column-wise zero fill) and bulk-stored
// with the CDNA5 Tensor Data Mover: one contiguous 2KB DMA per wave.
// ---------------------------------------------------------------------------

#define TPB 256
#define NSPAN 13  // interior knot spans: linspace(0, 1+1e-7, 14)

struct __align__(16) SpanTab {
  float kn[8];  // kn[q] = K[s+1+q], q=0..6  (K[m-2]..K[m+4], m=s+3); kn[7]=pad
  float rd[8];  // rd[0]=G1[m+1], rd[1..2]=G2[m+1..m+2], rd[3..5]=G3[m+1..m+3]
                // G[p][i] = 1 / (f32(K[i]-K[i-p]) + 1e-15)
};
struct Tables { SpanTab sp[NSPAN]; };

constexpr double knotd(int i) {
  // knots = [0,0,0, linspace(0, 1.0000001, 14) @ idx 3..16, 1,1,1]
  if (i < 3) return 0.0;
  if (i > 16) return 1.0;
  int j = i - 3;
  if (j == 13) return 1.0000001;  // linspace endpoint exact
  return (1.0000001 / 13.0) * (double)j;
}
constexpr float knotf(int i) { return (float)knotd(i); }

constexpr Tables make_tables() {
  Tables T{};
  for (int s = 0; s < NSPAN; ++s) {
    const int m = s + 3;
    for (int q = 0; q < 7; ++q) T.sp[s].kn[q] = knotf(s + 1 + q);
    T.sp[s].kn[7] = 0.0f;
    const int ps[6] = {1, 2, 2, 3, 3, 3};
    const int rs[6] = {1, 1, 2, 1, 2, 3};
    for (int e = 0; e < 6; ++e) {
      const int p = ps[e], r = rs[e];
      const float dn = (float)((double)knotf(m + r) - (double)knotf(m + r - p));
      T.sp[s].rd[e] = (float)(1.0 / ((double)dn + 1e-15));
    }
    T.sp[s].rd[6] = 0.0f;
    T.sp[s].rd[7] = 0.0f;
  }
  return T;
}

__constant__ Tables g_tab = make_tables();

typedef unsigned int v4u __attribute__((ext_vector_type(4)));
typedef int          v8i __attribute__((ext_vector_type(8)));
typedef int          v4i __attribute__((ext_vector_type(4)));

__global__ __launch_bounds__(TPB) void bspline16_kernel(
    const float* __restrict__ ts, float* __restrict__ out) {
  __shared__ float lds[TPB * 16];  // 16KB: one 16-float row per thread

  const int tid = threadIdx.x;
  const unsigned g = blockIdx.x * TPB + tid;
  const float t = ts[g];

  // ---- span index: s such that K[3+s] <= t < K[4+s], branchless -----------
  constexpr float INV13 = (float)(13.0 / 1.0000001);
  constexpr float HS    = (float)(1.0000001 / 13.0);
  int s = (int)(t * INV13);
  s = s < 0 ? 0 : (s > 12 ? 12 : s);
  s += (int)(t >= (float)(s + 1) * HS);  // conditions are mutually exclusive,
  s -= (int)(t < (float)s * HS);         // so sequential adjust == if/else-if
  s = s < 0 ? 0 : (s > 12 ? 12 : s);

  // ---- per-span constants (64B, hot in WGP$) ------------------------------
  const SpanTab* tb = &g_tab.sp[s];
  const float4 knA = ((const float4*)tb->kn)[0];
  const float4 knB = ((const float4*)tb->kn)[1];
  const float4 rdA = ((const float4*)tb->rd)[0];
  const float4 rdB = ((const float4*)tb->rd)[1];

  // ---- local de Boor: 4 nonzero basis values ------------------------------
  const float L1 = t - knA.y;   // t - K[m-1]
  const float L2 = t - knA.z;   // t - K[m]
  const float L3 = t - knA.w;   // t - K[m+1]
  const float R1 = knB.x - t;   // K[m+2] - t
  const float R2 = knB.y - t;   // K[m+3] - t
  const float R3 = knB.z - t;   // K[m+4] - t

  const float a0 = R1 * rdA.x;            // degree 1
  const float a1 = L3 * rdA.x;
  const float t0 = rdA.y * a0;            // degree 2
  const float t1 = rdA.z * a1;
  const float b0 = R1 * t0;
  const float b1 = fmaf(L2, t0, R2 * t1);
  const float b2 = L3 * t1;
  const float u0 = rdA.w * b0;            // degree 3
  const float u1 = rdB.x * b1;
  const float u2 = rdB.y * b2;
  const float c0 = R1 * u0;
  const float c1 = fmaf(L1, u0, R2 * u1);
  const float c2 = fmaf(L2, u1, R3 * u2);
  const float c3 = L3 * u2;

  // ---- zero the wave's 2KB LDS region, column-wise (conflict-free) --------
  // lane l writes float4 indices wave_base + c*32 + l: consecutive lanes hit
  // consecutive bank quads; stays wave-local so no __syncthreads is needed.
  const int lane = tid & 31;
  const int wave = tid >> 5;
  const float4 z = make_float4(0.f, 0.f, 0.f, 0.f);
  float4* l4 = (float4*)lds;
  const int wbase4 = wave * 128;  // 2KB region = 128 float4s
#pragma unroll
  for (int c = 0; c < 4; ++c) l4[wbase4 + c * 32 + lane] = z;

  // ---- scatter the 4 nonzero values into this thread's row ----------------
  float* rw = &lds[tid * 16 + s];
  rw[0] = c0; rw[1] = c1; rw[2] = c2; rw[3] = c3;

  // LDS writes must be visible to the TDM before it reads them.
  asm volatile("s_wait_dscnt 0" ::: "memory");

  // ---- one Tensor Data Mover store per wave: 2KB contiguous ---------------
  const unsigned lds_off =
      (unsigned)(uintptr_t)(&lds[0]) + (unsigned)(wave * 2048);
  const unsigned long long gbyte =
      (unsigned long long)(uintptr_t)out +
      (unsigned long long)blockIdx.x * (unsigned long long)(TPB * 16 * 4) +
      (unsigned long long)(wave * 2048);

#if __has_builtin(__builtin_amdgcn_tensor_store_from_lds)
  // D# group 0: count=1 | lds_addr | global_addr[31:0] | addr[56:32], type=2
  const unsigned w1 = (unsigned)__builtin_amdgcn_readfirstlane((int)lds_off);
  const unsigned w2 = (unsigned)__builtin_amdgcn_readfirstlane(
      (int)(unsigned)(gbyte & 0xffffffffull));
  const unsigned w3 = ((unsigned)__builtin_amdgcn_readfirstlane(
                           (int)(unsigned)((gbyte >> 32) & 0x1ffffffull))) |
                      (2u << 30);
  v4u g0 = {1u, w1, w2, w3};
  // D# group 1: data_size=4B; 1-D tile: tensor_dim0=tile_dim0=stride0=512 elts
  v8i g1 = {(int)0x00020000,     // workgroup_mask=0, data_size=2 (4B)
            (int)(512u << 16),   // tensor_dim0[15:0]
            (int)(1u << 16),     // tensor_dim1 = 1
            (int)(512u << 16),   // tile_dim0 = 512
            1,                   // tile_dim1 = 1, tile_dim2 = 0
            512,                 // tensor_dim0_stride[31:0]
            0, 0};
  v4i g2 = {0, 0, 0, 0};
  v4i g3 = {0, 0, 0, 0};
#if defined(__clang_major__) && __clang_major__ >= 23
  v8i g4 = {0, 0, 0, 0, 0, 0, 0, 0};
  __builtin_amdgcn_tensor_store_from_lds(g0, g1, g2, g3, g4, 0);
#else
  __builtin_amdgcn_tensor_store_from_lds(g0, g1, g2, g3, 0);
#endif
  __builtin_amdgcn_s_wait_tensorcnt(0);
#else
  // Fallback: coalesced LDS -> global copy (b128 per lane, unit stride).
  __syncthreads();
  const float4* src = (const float4*)lds;
  float4* dst = (float4*)(out + (unsigned long long)blockIdx.x * (TPB * 16));
#pragma unroll
  for (int k = 0; k < 4; ++k) dst[tid + k * TPB] = src[tid + k * TPB];
#endif
}

extern "C" void kernel_launch(void* const* d_in, const int* in_sizes, int n_in,
                              void* d_out, int out_size, void* d_ws, size_t ws_size,
                              hipStream_t stream) {
  const float* ts = (const float*)d_in[0];
  float* out = (float*)d_out;
  const int n = in_sizes[0];                 // 32*131072 = 4,194,304 (mult of 256)
  const int blocks = (n + TPB - 1) / TPB;    // 16384
  bspline16_kernel<<<blocks, TPB, 0, stream>>>(ts, out);
}